// AttentionLayer_61641370632996
// MI455X (gfx1250) — compile-verified
//
#include <hip/hip_runtime.h>
#include <math.h>

typedef __attribute__((ext_vector_type(2))) float v2f;
typedef __attribute__((ext_vector_type(4))) float v4f;
typedef __attribute__((ext_vector_type(8))) float v8f;
typedef __attribute__((ext_vector_type(4))) int   v4i;

#define AS1 __attribute__((address_space(1)))
#define AS3 __attribute__((address_space(3)))

#if __has_builtin(__builtin_amdgcn_global_load_async_to_lds_b128)
#define HAVE_ASYNC_LDS 1
#endif

#define B_ 32
#define T_ 2048
#define J_ 128
#define D_ 256

// LDS strides (floats), chosen for bank-conflict-free WMMA fragment feeds
#define US 268   // U tile row stride    (268 % 64 == 12)
#define SS 132   // S-stage row stride   (132 % 64 == 4)

// LDS float offsets
#define OFF_SU 0
#define OFF_W1 (128 * US)            // 34304
#define OFF_W3 (OFF_W1 + 256)
#define OFF_S2 (OFF_W3 + 256)
#define OFF_RM (OFF_S2 + 128)        // [8 waves][16 rows][max,inv]
#define OFF_SS (OFF_RM + 256)
#define LDS_FLOATS (OFF_SS + 8 * 16 * SS)   // 52096 floats = 208384 B

// ---------------------------------------------------------------- s2 = U @ w2
__global__ void k_s2(const float* __restrict__ U, const float* __restrict__ Ws,
                     float* __restrict__ s2) {
  int row  = blockIdx.x * 4 + (threadIdx.x >> 5);  // one wave per (b,j) row
  int lane = threadIdx.x & 31;
  const float* w2 = Ws + 256;
  const float* ur = U + (size_t)row * D_;
  float acc = 0.f;
  for (int i = lane; i < D_; i += 32) acc = fmaf(ur[i], w2[i], acc);
  for (int off = 16; off > 0; off >>= 1) acc += __shfl_xor(acc, off, 32);
  if (lane == 0) s2[row] = acc;
}

// ------------------------------------------------- main: S, softmax_j, AQ, G0-2
__global__ __launch_bounds__(256, 1)
void k_bidaf_main(const float* __restrict__ H, const float* __restrict__ U,
                  const float* __restrict__ Ws, const float* __restrict__ s2g,
                  float* __restrict__ G, float* __restrict__ S,
                  float* __restrict__ rowmax_g) {
  extern __shared__ float lds[];
  float* sU    = lds + OFF_SU;
  float* sW1   = lds + OFF_W1;
  float* sW3   = lds + OFF_W3;
  float* s2s   = lds + OFF_S2;
  float* rmInv = lds + OFF_RM;
  float* sS    = lds + OFF_SS;

  const int tid  = threadIdx.x;
  const int wave = tid >> 5;
  const int lane = tid & 31;
  const int m16  = lane & 15;
  const int hi   = lane >> 4;
  const int kb   = hi * 2;           // K-pair base per WMMA f32 A/B layout
  const int b    = blockIdx.y;
  const int t0   = blockIdx.x * 128 + wave * 16;

  // Stage U[b] (128x256 f32) into padded LDS; also w1/w3/s2.
  {
    const float* Ub = U + (size_t)b * (J_ * D_);
#ifdef HAVE_ASYNC_LDS
    // CDNA5 async global->LDS DMA: no VGPR round-trip, tracked by ASYNCcnt.
    for (int i = tid; i < (J_ * D_) / 4; i += 256) {
      int j  = i >> 6;
      int d4 = (i & 63) << 2;
      __builtin_amdgcn_global_load_async_to_lds_b128(
          (AS1 v4i*)(Ub + ((size_t)i << 2)),
          (AS3 v4i*)(sU + j * US + d4), 0, 0);
    }
#if __has_builtin(__builtin_amdgcn_s_wait_asynccnt)
    __builtin_amdgcn_s_wait_asynccnt(0);
#else
    asm volatile("s_wait_asynccnt 0" ::: "memory");
#endif
#else
    for (int i = tid; i < (J_ * D_) / 4; i += 256) {
      int j  = i >> 6;
      int d4 = (i & 63) << 2;
      v4f v = *(const v4f*)(Ub + ((size_t)i << 2));
      *(v4f*)(sU + j * US + d4) = v;
    }
#endif
    sW1[tid] = Ws[tid];
    sW3[tid] = Ws[512 + tid];
    if (tid < 128) s2s[tid] = s2g[b * J_ + tid];
  }
  __syncthreads();

  // ---- GEMM1: S' = H @ (w3*U^T + w1)   [16 x 128 per wave, K = 256]
  v8f acc[8] = {};
  const float* Hrow = H + ((size_t)b * T_ + t0 + m16) * D_;
  for (int kk = 0; kk < D_; kk += 4) {
    v2f a   = *(const v2f*)(Hrow + kk + kb);
    v2f w3v = *(const v2f*)(sW3 + kk + kb);
    v2f w1v = *(const v2f*)(sW1 + kk + kb);
#pragma unroll
    for (int nt = 0; nt < 8; ++nt) {
      v2f u = *(const v2f*)(sU + (nt * 16 + m16) * US + kk + kb);
      v2f bf;
      bf.x = w3v.x * u.x + w1v.x;
      bf.y = w3v.y * u.y + w1v.y;
      acc[nt] = __builtin_amdgcn_wmma_f32_16x16x4_f32(
          false, a, false, bf, (short)0, acc[nt], false, false);
    }
  }

  // ---- Add s2, stage tile to LDS, stream S out (non-temporal)
  float* sSw = sS + wave * (16 * SS);
  const size_t Sbase = ((size_t)b * T_ + t0) * J_;
#pragma unroll
  for (int nt = 0; nt < 8; ++nt) {
    float s2v = s2s[nt * 16 + m16];
#pragma unroll
    for (int r = 0; r < 8; ++r) {
      int m   = r + 8 * hi;
      float v = acc[nt][r] + s2v;
      sSw[m * SS + nt * 16 + m16] = v;
      __builtin_nontemporal_store(v, S + Sbase + (size_t)m * J_ + nt * 16 + m16);
    }
  }

  // ---- Row softmax stats over J=128 (2 lanes per row, 64 cols each)
  {
    const float* rp = sSw + m16 * SS + hi * 64;
    float mx = -__builtin_inff();
#pragma unroll
    for (int j = 0; j < 64; j += 4) {
      v4f v = *(const v4f*)(rp + j);
      mx = fmaxf(mx, fmaxf(fmaxf(v.x, v.y), fmaxf(v.z, v.w)));
    }
    mx = fmaxf(mx, __shfl_xor(mx, 16, 32));
    float sum = 0.f;
#pragma unroll
    for (int j = 0; j < 64; j += 4) {
      v4f v = *(const v4f*)(rp + j);
      sum += __expf(v.x - mx) + __expf(v.y - mx) + __expf(v.z - mx) + __expf(v.w - mx);
    }
    sum += __shfl_xor(sum, 16, 32);
    if (hi == 0) {
      rmInv[(wave * 16 + m16) * 2 + 0] = mx;
      rmInv[(wave * 16 + m16) * 2 + 1] = 1.f / sum;
      rowmax_g[(size_t)b * T_ + t0 + m16] = mx;
    }
  }
  float rm = rmInv[(wave * 16 + m16) * 2 + 0];
  float ri = rmInv[(wave * 16 + m16) * 2 + 1];

  // ---- GEMM2: AQ = softmax_j(S) @ U   [K = 128, N = 256 in two halves]
  const size_t HbaseT = ((size_t)b * T_ + t0) * D_;
  const size_t Gbase  = ((size_t)b * T_ + t0) * (4 * D_);
#pragma unroll
  for (int nh = 0; nh < 2; ++nh) {
    const int d0b = nh * 128;
    v8f acc2[8] = {};
    for (int kj = 0; kj < J_; kj += 4) {
      v2f sv = *(const v2f*)(sSw + m16 * SS + kj + kb);
      v2f af;
      af.x = __expf(sv.x - rm) * ri;
      af.y = __expf(sv.y - rm) * ri;
#pragma unroll
      for (int nt = 0; nt < 8; ++nt) {
        int d0 = d0b + nt * 16;
        v2f bf;
        bf.x = sU[(kj + kb) * US + d0 + m16];
        bf.y = sU[(kj + kb + 1) * US + d0 + m16];
        acc2[nt] = __builtin_amdgcn_wmma_f32_16x16x4_f32(
            false, af, false, bf, (short)0, acc2[nt], false, false);
      }
    }
    // G0 = H, G1 = AQ, G2 = H*AQ for this d-half (NT streaming stores)
#pragma unroll
    for (int nt = 0; nt < 8; ++nt) {
      int d = d0b + nt * 16 + m16;
#pragma unroll
      for (int r = 0; r < 8; ++r) {
        int m      = r + 8 * hi;
        float hval = H[HbaseT + (size_t)m * D_ + d];   // L2-hot reload
        float aq   = acc2[nt][r];
        float* gp  = G + Gbase + (size_t)m * (4 * D_);
        __builtin_nontemporal_store(hval,        gp + d);
        __builtin_nontemporal_store(aq,          gp + D_ + d);
        __builtin_nontemporal_store(hval * aq,   gp + 2 * D_ + d);
      }
    }
  }
}

// ---------------------------------------- q2c: softmax over t of rowmax
__global__ void k_softmax_t(const float* __restrict__ rowmax_g, float* __restrict__ p) {
  __shared__ float red[256];
  int b = blockIdx.x, tid = threadIdx.x;
  const float* mrow = rowmax_g + (size_t)b * T_;
  float mx = -__builtin_inff();
  for (int t = tid; t < T_; t += 256) mx = fmaxf(mx, mrow[t]);
  red[tid] = mx; __syncthreads();
  for (int s = 128; s > 0; s >>= 1) { if (tid < s) red[tid] = fmaxf(red[tid], red[tid + s]); __syncthreads(); }
  mx = red[0]; __syncthreads();
  float sum = 0.f;
  for (int t = tid; t < T_; t += 256) sum += __expf(mrow[t] - mx);
  red[tid] = sum; __syncthreads();
  for (int s = 128; s > 0; s >>= 1) { if (tid < s) red[tid] += red[tid + s]; __syncthreads(); }
  float inv = 1.f / red[0];
  float* pb = p + (size_t)b * T_;
  for (int t = tid; t < T_; t += 256) pb[t] = __expf(mrow[t] - mx) * inv;
}

// h partials over 128-row t-chunks (parallel BW, no float atomics -> deterministic)
#define TCH 128
__global__ void k_hpart(const float* __restrict__ H, const float* __restrict__ p,
                        float* __restrict__ hpart) {
  int b = blockIdx.x, c = blockIdx.y, d = threadIdx.x;
  __shared__ float pc[TCH];
  if (d < TCH) pc[d] = p[(size_t)b * T_ + c * TCH + d];
  __syncthreads();
  const float* Hb = H + ((size_t)b * T_ + (size_t)c * TCH) * D_;
  float acc = 0.f;
  for (int t = 0; t < TCH; ++t) acc = fmaf(pc[t], Hb[(size_t)t * D_ + d], acc);
  hpart[((size_t)b * 16 + c) * D_ + d] = acc;
}

__global__ void k_hfold(const float* __restrict__ hpart, float* __restrict__ hvec) {
  int b = blockIdx.x, d = threadIdx.x;
  float acc = 0.f;
#pragma unroll
  for (int c = 0; c < 16; ++c) acc += hpart[((size_t)b * 16 + c) * D_ + d];
  hvec[(size_t)b * D_ + d] = acc;
}

// ---------------------------------------- G3 = H * h  (float4 NT streaming)
__global__ void k_g3(const float* __restrict__ H, const float* __restrict__ hvec,
                     float* __restrict__ G) {
  int i   = blockIdx.x * blockDim.x + threadIdx.x;  // float4 index
  int d4  = (i & 63) << 2;
  int row = i >> 6;                                  // b*T + t
  int b   = row >> 11;
  v4f hv = *(const v4f*)(H + ((size_t)row << 8) + d4);
  v4f hh = *(const v4f*)(hvec + ((size_t)b << 8) + d4);
  v4f g  = hv * hh;
  __builtin_nontemporal_store(g, (v4f*)(G + (size_t)row * 1024 + 768 + d4));
}

extern "C" void kernel_launch(void* const* d_in, const int* in_sizes, int n_in,
                              void* d_out, int out_size, void* d_ws, size_t ws_size,
                              hipStream_t stream) {
  (void)in_sizes; (void)n_in; (void)out_size; (void)ws_size;
  const float* H  = (const float*)d_in[0];
  const float* U  = (const float*)d_in[1];
  const float* Ws = (const float*)d_in[2];

  float* out = (float*)d_out;
  float* G = out;                                   // (B,T,1024)
  float* S = out + (size_t)B_ * T_ * (4 * D_);      // (B,T,J)

  float* w      = (float*)d_ws;
  float* s2     = w;                  // B*J      = 4096
  float* rowmax = s2 + B_ * J_;       // B*T      = 65536
  float* p      = rowmax + (size_t)B_ * T_;   // B*T
  float* hpart  = p + (size_t)B_ * T_;        // B*16*D = 131072
  float* hvec   = hpart + (size_t)B_ * 16 * D_;  // B*D = 8192

  k_s2<<<dim3((B_ * J_) / 4), dim3(128), 0, stream>>>(U, Ws, s2);

  size_t ldsBytes = (size_t)LDS_FLOATS * sizeof(float);
  (void)hipFuncSetAttribute((const void*)k_bidaf_main,
                            hipFuncAttributeMaxDynamicSharedMemorySize,
                            (int)ldsBytes);
  k_bidaf_main<<<dim3(T_ / 128, B_), dim3(256), ldsBytes, stream>>>(
      H, U, Ws, s2, G, S, rowmax);

  k_softmax_t<<<dim3(B_), dim3(256), 0, stream>>>(rowmax, p);
  k_hpart<<<dim3(B_, 16), dim3(256), 0, stream>>>(H, p, hpart);
  k_hfold<<<dim3(B_), dim3(256), 0, stream>>>(hpart, hvec);
  k_g3<<<dim3((B_ * T_ * D_ / 4) / 256), dim3(256), 0, stream>>>(H, hvec, G);
}